// TinyRNN_19559281066344
// MI455X (gfx1250) — compile-verified
//
#include <hip/hip_runtime.h>

// TinyGRU scan for MI455X (gfx1250): fp32 WMMA recurrence, one workgroup per
// 16-batch tile owns the whole T=2048 scan. D = A(16x4) * B(4x16) + C(16x16)
// via V_WMMA_F32_16X16X4_F32; K = 36 = 32 (h @ W_hh^T) + 4 ([x,0] @ W_ih^T).

typedef __attribute__((ext_vector_type(2))) float v2f;
typedef __attribute__((ext_vector_type(8))) float v8f;

#define IN_DIM 3
#define HID    32
#define TLEN   2048
#define BATCH  1024

__device__ __forceinline__ float fast_sigmoid(float x) {
    return __builtin_amdgcn_rcpf(1.0f + __expf(-x));
}
__device__ __forceinline__ float fast_tanh(float x) {
    // tanh(x) = 2*sigmoid(2x) - 1
    return __builtin_fmaf(2.0f, fast_sigmoid(2.0f * x), -1.0f);
}

__device__ __forceinline__ v8f wmma4(v2f a, v2f b, v8f c) {
    return __builtin_amdgcn_wmma_f32_16x16x4_f32(
        /*neg_a=*/false, a, /*neg_b=*/false, b,
        /*c_mod=*/(short)0, c, /*reuse_a=*/false, /*reuse_b=*/false);
}

__global__ __launch_bounds__(64) void tiny_gru_scan(
    const float* __restrict__ x,     // [B, T, 3]
    const float* __restrict__ W_ih,  // [96, 3]
    const float* __restrict__ W_hh,  // [96, 32]
    const float* __restrict__ b_ih,  // [96]
    const float* __restrict__ b_hh,  // [96]
    const float* __restrict__ W_fc,  // [3, 32]
    const float* __restrict__ b_fc,  // [3]
    float* __restrict__ out)         // [B, 3]
{
    // Double-buffered h tile: 16 batch rows x 32 hidden, padded row stride 33
    // (64-bank LDS, 33 coprime with 64 -> conflict-light store/load patterns).
    __shared__ float hbuf[2][16 * 33];

    const int tid  = threadIdx.x;
    const int wv   = tid >> 5;        // wave 0 -> hidden j in [0,16), wave 1 -> [16,32)
    const int lane = tid & 31;
    const int L16  = lane & 15;       // lane within half-wave
    const int hi   = lane >> 4;       // 0: holds K pair {4c,4c+1}; 1: {4c+2,4c+3}
    const int B0   = blockIdx.x * 16; // batch tile base

    // h0 = 0
    for (int i = tid; i < 2 * 16 * 33; i += 64) ((float*)hbuf)[i] = 0.0f;

    // ---- Preload B-matrix fragments (resident in VGPRs across the scan) ----
    // B tile layout (4x16 f32): v0 = row K (lo lanes K=4c, hi lanes K=4c+2),
    // v1 = next row; N = gate column = lane%16 within tile.
    const int gr = wv * 16 + L16;       // reset-gate column
    const int gz = 32 + wv * 16 + L16;  // update-gate column
    const int gn = 64 + wv * 16 + L16;  // new-gate column

    v2f bR[9], bZ[9], bN[9];
#pragma unroll
    for (int c = 0; c < 8; ++c) {       // K rows 0..31 come from W_hh^T
        const int k0 = 4 * c + 2 * hi;
        bR[c].x = W_hh[gr * HID + k0]; bR[c].y = W_hh[gr * HID + k0 + 1];
        bZ[c].x = W_hh[gz * HID + k0]; bZ[c].y = W_hh[gz * HID + k0 + 1];
        bN[c].x = W_hh[gn * HID + k0]; bN[c].y = W_hh[gn * HID + k0 + 1];
    }
    // K rows 32..35 = [W_ih^T row0, row1, row2, zero-pad]
    {
        float r0 = W_ih[gr * IN_DIM + 2 * hi], r1 = W_ih[gr * IN_DIM + 1];
        float z0 = W_ih[gz * IN_DIM + 2 * hi], z1 = W_ih[gz * IN_DIM + 1];
        float n0 = W_ih[gn * IN_DIM + 2 * hi], n1 = W_ih[gn * IN_DIM + 1];
        bR[8].x = r0; bR[8].y = hi ? 0.0f : r1;
        bZ[8].x = z0; bZ[8].y = hi ? 0.0f : z1;
        bN[8].x = n0; bN[8].y = hi ? 0.0f : n1;
    }

    // Biases folded into C (broadcast along batch rows == all 8 acc VGPRs).
    const float cR  = b_ih[gr] + b_hh[gr];
    const float cZ  = b_ih[gz] + b_hh[gz];
    const float cNh = b_hh[gn];   // r multiplies (h@W_hhn + b_hhn) only
    const float cNx = b_ih[gn];

    float hprev[8];
#pragma unroll
    for (int v = 0; v < 8; ++v) hprev[v] = 0.0f;

    const float* xrow = x + (size_t)(B0 + L16) * TLEN * IN_DIM;
    const int myj = wv * 16 + L16;   // hidden unit this lane produces

    __syncthreads();

#pragma unroll 1
    for (int t = 0; t < TLEN; ++t) {
        const float* cur = hbuf[t & 1];
        float*       nxt = hbuf[(t + 1) & 1];

        // A fragments: 16x4 f32 — lane = batch row, v0/v1 = two K values.
        v2f a[9];
#pragma unroll
        for (int c = 0; c < 8; ++c) {
            const float* p = cur + L16 * 33 + 4 * c + 2 * hi;
            a[c].x = p[0]; a[c].y = p[1];
        }
        {   // K chunk 8: x_t rows (lo lanes: x0,x1; hi lanes: x2,0)
            const float* xp = xrow + t * IN_DIM;
            float x0 = xp[2 * hi];
            float x1 = xp[1];
            a[8].x = x0; a[8].y = hi ? 0.0f : x1;
        }

        v8f accR  = { cR,  cR,  cR,  cR,  cR,  cR,  cR,  cR  };
        v8f accZ  = { cZ,  cZ,  cZ,  cZ,  cZ,  cZ,  cZ,  cZ  };
        v8f accNh = { cNh, cNh, cNh, cNh, cNh, cNh, cNh, cNh };
        v8f accNx = { cNx, cNx, cNx, cNx, cNx, cNx, cNx, cNx };

#pragma unroll
        for (int c = 0; c < 9; ++c) {
            accR = wmma4(a[c], bR[c], accR);
            accZ = wmma4(a[c], bZ[c], accZ);
        }
#pragma unroll
        for (int c = 0; c < 8; ++c)       // hn = h @ W_hhn^T + b_hhn
            accNh = wmma4(a[c], bN[c], accNh);
        accNx = wmma4(a[8], bN[8], accNx); // xn = x @ W_ihn^T + b_ihn

        // Gates: in D layout lane = gate column, VGPR v = batch row v+8*hi,
        // so r/z/hn/xn for a given (batch, j) are co-resident per lane slot.
#pragma unroll
        for (int v = 0; v < 8; ++v) {
            float r = fast_sigmoid(accR[v]);
            float z = fast_sigmoid(accZ[v]);
            float n = fast_tanh(__builtin_fmaf(r, accNh[v], accNx[v]));
            float h = __builtin_fmaf(z, hprev[v] - n, n);  // (1-z)n + z*h
            hprev[v] = h;
            nxt[(v + 8 * hi) * 33 + myj] = h;
        }
        __syncthreads();
    }

    // Final projection: out = hT @ W_fc^T + b_fc  (hT sits in hbuf[0], T even)
    if (tid < 16) {
        const float* hT = hbuf[0] + tid * 33;
#pragma unroll
        for (int g = 0; g < 3; ++g) {
            float s = b_fc[g];
#pragma unroll
            for (int j = 0; j < HID; ++j)
                s = __builtin_fmaf(hT[j], W_fc[g * HID + j], s);
            out[(B0 + tid) * 3 + g] = s;
        }
    }
}

extern "C" void kernel_launch(void* const* d_in, const int* in_sizes, int n_in,
                              void* d_out, int out_size, void* d_ws, size_t ws_size,
                              hipStream_t stream) {
    (void)in_sizes; (void)n_in; (void)d_ws; (void)ws_size; (void)out_size;
    const float* x    = (const float*)d_in[0];
    const float* W_ih = (const float*)d_in[1];
    const float* W_hh = (const float*)d_in[2];
    const float* b_ih = (const float*)d_in[3];
    const float* b_hh = (const float*)d_in[4];
    const float* W_fc = (const float*)d_in[5];
    const float* b_fc = (const float*)d_in[6];
    hipLaunchKernelGGL(tiny_gru_scan, dim3(BATCH / 16), dim3(64), 0, stream,
                       x, W_ih, W_hh, b_ih, b_hh, W_fc, b_fc, (float*)d_out);
}